// Basic_GAT_80874234184376
// MI455X (gfx1250) — compile-verified
//
#include <hip/hip_runtime.h>
#include <hip/hip_bf16.h>

typedef float v2f __attribute__((ext_vector_type(2)));
typedef float v8f __attribute__((ext_vector_type(8)));

#define BN 1024
#define FNDIM 128
#define OUTDIM 256
#define NH 16

// ---------------------------------------------------------------------------
// Kernel C: att_g[b,h] = graph_fts[b,:] @ ag_w + ag_b   (trivial)
// ---------------------------------------------------------------------------
__global__ void attg_kernel(const float* __restrict__ graph,
                            const float* __restrict__ ag_w,
                            const float* __restrict__ ag_b,
                            float* __restrict__ attg) {
    int t = threadIdx.x;
    if (t >= 64) return;
    int b = t >> 4, h = t & 15;
    float s = ag_b[h];
    for (int k = 0; k < 128; ++k) s += graph[b * 128 + k] * ag_w[k * 16 + h];
    attg[t] = s;
}

// ---------------------------------------------------------------------------
// Kernel A: node projections via V_WMMA_F32_16X16X4_F32.
// One block = (b, 16-row tile). 16 waves. 34 16-col output tiles.
// All indexing kept in 32-bit off uniform bases -> saddr+voffset addressing.
// ---------------------------------------------------------------------------
__global__ __launch_bounds__(512, 2) void proj_kernel(
    const float* __restrict__ node,
    const float* __restrict__ m_w,   const float* __restrict__ m_b,
    const float* __restrict__ sk_w,  const float* __restrict__ sk_b,
    const float* __restrict__ a1_w,  const float* __restrict__ a1_b,
    const float* __restrict__ a2_w,  const float* __restrict__ a2_b,
    float* __restrict__ values, float* __restrict__ skipo,
    float* __restrict__ att1,   float* __restrict__ att2) {
    const int blk  = blockIdx.x;
    const int b    = blk >> 6;
    const int i0   = (blk & 63) << 4;
    const int w    = threadIdx.x >> 5;
    const int l    = threadIdx.x & 31;
    const int hl   = l & 15;
    const int half = l >> 4;

    const float* Abase = node + (b * BN + i0) * FNDIM;  // uniform base
    const int arow = hl * FNDIM;                        // lane offset (32-bit)

    for (int t = w; t < 34; t += 16) {
        const float* W; const float* bias; float* Out; int ncol, colbase;
        if (t < 16)       { W = m_w;  bias = m_b;  Out = values; ncol = OUTDIM; colbase = t * 16; }
        else if (t < 32)  { W = sk_w; bias = sk_b; Out = skipo;  ncol = OUTDIM; colbase = (t - 16) * 16; }
        else if (t == 32) { W = a1_w; bias = a1_b; Out = att1;   ncol = NH;     colbase = 0; }
        else              { W = a2_w; bias = a2_b; Out = att2;   ncol = NH;     colbase = 0; }
        const int col = colbase + hl;
        float* outb = Out + b * BN * ncol;              // uniform base per batch
        v8f c = {};
        for (int kc = 0; kc < 32; ++kc) {
            const int k = kc * 4 + half * 2;
            v2f a = *(const v2f*)(Abase + arow + k);
            v2f bb;
            bb.x = W[k * ncol + col];
            bb.y = W[(k + 1) * ncol + col];
            c = __builtin_amdgcn_wmma_f32_16x16x4_f32(false, a, false, bb,
                                                      (short)0, c, false, false);
        }
        const float bv = bias[col];
#pragma unroll
        for (int r = 0; r < 8; ++r) {
            const int row = i0 + r + half * 8;
            outb[row * ncol + col] = c[r] + bv;
        }
    }
}

// ---------------------------------------------------------------------------
// Fused GAT kernel. One block = (b, 16-row i-tile). 16 waves = 512 threads.
// Edge tiles (contiguous 1KB) are double-buffered into LDS with
// GLOBAL_LOAD_ASYNC_TO_LDS_B128 (ASYNCcnt), hiding HBM latency:
//   issue tile jt+1 -> s_wait_asynccnt 2 -> consume tile jt.
// Per j-tile:
//   stage 1 (wave w == row i0+w): E = edge[16j x 16k] @ ae_w  (4x WMMA f32),
//            logits -> leaky_relu -> +adj bias -> exp -> P_lds[h][i][j]
//   stage 2 (wave w == head w):   acc += P[16i x 16j] @ V[16j x 16hd] (4x WMMA)
// Then row-sum normalization (ds_swizzle xor16), skip+relu, LayerNorm.
// ---------------------------------------------------------------------------
__global__ __launch_bounds__(512, 2) void gat_fused(
    const float* __restrict__ edge, const int* __restrict__ adj,
    const float* __restrict__ ae_w, const float* __restrict__ ae_b,
    const float* __restrict__ values, const float* __restrict__ skipv,
    const float* __restrict__ att1, const float* __restrict__ att2,
    const float* __restrict__ attg,
    const float* __restrict__ ln_scale, const float* __restrict__ ln_off,
    float* __restrict__ out) {
    __shared__ float edge_sh[16 * 512];  // per wave: 2 x 256-float tile buffers
    __shared__ float P_lds[16 * 281];    // [h]*281 + [i]*17 + [j]
    __shared__ float ln_buf[16 * 257];   // [i]*257 + [out]
    __shared__ float s_lds[16 * 17];     // [i]*17 + [h]

    const int blk  = blockIdx.x;
    const int b    = blk >> 6;
    const int i0   = (blk & 63) << 4;
    const int w    = threadIdx.x >> 5;   // wave id: row (stage1) / head (stage2)
    const int l    = threadIdx.x & 31;
    const int hl   = l & 15;
    const int half = l >> 4;
    const int i    = i0 + w;

    // Uniform per-batch base pointers (saddr), 32-bit lane offsets.
    const float* att2b = att2 + b * BN * NH;
    const float* valb  = values + b * BN * OUTDIM;
    const float* skipb = skipv + b * BN * OUTDIM;
    const float* erow  = edge + ((size_t)(b * BN) + i) * BN * 16;  // wave-uniform
    const int*   adjrow = adj + (b * BN + i) * BN;

    // Loop-invariant B operand (ae_w), 4 K-chunks in WMMA B layout.
    v2f Bae[4];
#pragma unroll
    for (int c = 0; c < 4; ++c) {
        const int k = c * 4 + half * 2;
        Bae[c].x = ae_w[k * NH + hl];
        Bae[c].y = ae_w[(k + 1) * NH + hl];
    }
    const float base1 = att1[(b * BN + i) * NH + hl] + attg[b * NH + hl] + ae_b[hl];

    float* mybuf = &edge_sh[w * 512];
    const unsigned lds0 = (unsigned)(uintptr_t)(void*)mybuf;  // LDS byte offset

    // Issue async copy of edge tile jt into buffer (jt&1): 1KB via 2x B128.
    auto issue_tile = [&](int jt) {
        const unsigned ldsa = lds0 + (unsigned)((jt & 1) * 1024 + l * 16);
        const unsigned long long ga =
            (unsigned long long)(uintptr_t)(erow + jt * 256) + (unsigned)(l * 16);
        asm volatile("global_load_async_to_lds_b128 %0, %1, off"
                     :: "v"(ldsa), "v"(ga) : "memory");
        asm volatile("global_load_async_to_lds_b128 %0, %1, off offset:512"
                     :: "v"(ldsa), "v"(ga) : "memory");
    };

    v8f acc = {};
    float sumP = 0.f;

    issue_tile(0);
    for (int jt = 0; jt < 64; ++jt) {
        const int j0 = jt * 16;
        if (jt < 63) {
            issue_tile(jt + 1);
            asm volatile("s_wait_asynccnt 0x2" ::: "memory");
        } else {
            asm volatile("s_wait_asynccnt 0x0" ::: "memory");
        }
        const float* tile = mybuf + (jt & 1) * 256;

        // -------- stage 1: edge projection tile for row i --------
        v8f E = {};
#pragma unroll
        for (int c = 0; c < 4; ++c) {
            v2f a = *(const v2f*)(tile + hl * 16 + c * 4 + half * 2);
            E = __builtin_amdgcn_wmma_f32_16x16x4_f32(false, a, false, Bae[c],
                                                      (short)0, E, false, false);
        }

#pragma unroll
        for (int r = 0; r < 8; ++r) {
            const int jl = r + half * 8;
            const int jj = j0 + jl;
            float x = E[r] + base1 + att2b[jj * NH + hl];
            x = x > 0.f ? x : 0.01f * x;              // leaky_relu
            if (adjrow[jj] == 0) x -= 1e9f;           // adjacency bias
            const float p = __expf(x);                // max-free softmax numerator
            sumP += p;
            P_lds[hl * 281 + w * 17 + jl] = p;
        }
        __syncthreads();

        // -------- stage 2: head w accumulates P @ V --------
#pragma unroll
        for (int c = 0; c < 4; ++c) {
            const int k = c * 4 + half * 2;
            v2f a;
            a.x = P_lds[w * 281 + hl * 17 + k];
            a.y = P_lds[w * 281 + hl * 17 + k + 1];
            v2f bb;
            bb.x = valb[(j0 + k) * OUTDIM + w * 16 + hl];
            bb.y = valb[(j0 + k + 1) * OUTDIM + w * 16 + hl];
            acc = __builtin_amdgcn_wmma_f32_16x16x4_f32(false, a, false, bb,
                                                        (short)0, acc, false, false);
        }
        __syncthreads();
    }

    // -------- softmax denominators: combine lane halves (xor 16) --------
    {
        const int so = __builtin_amdgcn_ds_swizzle(__float_as_int(sumP), 0x401F);
        sumP += __int_as_float(so);
        if (half == 0) s_lds[w * 17 + hl] = sumP;
    }
    __syncthreads();

    // -------- normalize, add skip, ReLU -> ln_buf --------
    {
        const int outc = w * 16 + hl;
#pragma unroll
        for (int r = 0; r < 8; ++r) {
            const int il = r + half * 8;
            float v = acc[r] / s_lds[il * 17 + w];
            v += skipb[(i0 + il) * OUTDIM + outc];
            v = v > 0.f ? v : 0.f;
            ln_buf[il * 257 + outc] = v;
        }
    }
    __syncthreads();

    // -------- LayerNorm: wave w handles row i0+w --------
    {
        float vals[8];
        float s1 = 0.f, s2 = 0.f;
#pragma unroll
        for (int t = 0; t < 8; ++t) {
            const float v = ln_buf[w * 257 + l + 32 * t];
            vals[t] = v; s1 += v; s2 += v * v;
        }
        s1 += __int_as_float(__builtin_amdgcn_ds_swizzle(__float_as_int(s1), 0x041F));
        s2 += __int_as_float(__builtin_amdgcn_ds_swizzle(__float_as_int(s2), 0x041F));
        s1 += __int_as_float(__builtin_amdgcn_ds_swizzle(__float_as_int(s1), 0x081F));
        s2 += __int_as_float(__builtin_amdgcn_ds_swizzle(__float_as_int(s2), 0x081F));
        s1 += __int_as_float(__builtin_amdgcn_ds_swizzle(__float_as_int(s1), 0x101F));
        s2 += __int_as_float(__builtin_amdgcn_ds_swizzle(__float_as_int(s2), 0x101F));
        s1 += __int_as_float(__builtin_amdgcn_ds_swizzle(__float_as_int(s1), 0x201F));
        s2 += __int_as_float(__builtin_amdgcn_ds_swizzle(__float_as_int(s2), 0x201F));
        s1 += __int_as_float(__builtin_amdgcn_ds_swizzle(__float_as_int(s1), 0x401F));
        s2 += __int_as_float(__builtin_amdgcn_ds_swizzle(__float_as_int(s2), 0x401F));
        const float mean = s1 * (1.f / 256.f);
        const float var  = s2 * (1.f / 256.f) - mean * mean;
        const float rs   = rsqrtf(var + 1e-5f);
        float* orow = out + ((size_t)(b * BN) + i0 + w) * OUTDIM;
#pragma unroll
        for (int t = 0; t < 8; ++t) {
            const int idx = l + 32 * t;
            orow[idx] = (vals[t] - mean) * rs * ln_scale[idx] + ln_off[idx];
        }
    }
}

// ---------------------------------------------------------------------------
extern "C" void kernel_launch(void* const* d_in, const int* in_sizes, int n_in,
                              void* d_out, int out_size, void* d_ws, size_t ws_size,
                              hipStream_t stream) {
    (void)in_sizes; (void)n_in; (void)out_size; (void)ws_size;
    const float* node  = (const float*)d_in[0];
    const float* edge  = (const float*)d_in[1];
    const float* graph = (const float*)d_in[2];
    const int*   adj   = (const int*)d_in[3];
    // d_in[4] = hidden (unused by reference)
    const float* m_w   = (const float*)d_in[5];
    const float* m_b   = (const float*)d_in[6];
    const float* sk_w  = (const float*)d_in[7];
    const float* sk_b  = (const float*)d_in[8];
    const float* a1_w  = (const float*)d_in[9];
    const float* a1_b  = (const float*)d_in[10];
    const float* a2_w  = (const float*)d_in[11];
    const float* a2_b  = (const float*)d_in[12];
    const float* ae_w  = (const float*)d_in[13];
    const float* ae_b  = (const float*)d_in[14];
    const float* ag_w  = (const float*)d_in[15];
    const float* ag_b  = (const float*)d_in[16];
    const float* ln_s  = (const float*)d_in[17];
    const float* ln_o  = (const float*)d_in[18];
    float* out = (float*)d_out;

    float* ws     = (float*)d_ws;
    float* values = ws;                                  // 4*1024*256
    float* skipv  = values + (size_t)4 * BN * OUTDIM;    // 4*1024*256
    float* att1   = skipv  + (size_t)4 * BN * OUTDIM;    // 4*1024*16
    float* att2   = att1   + (size_t)4 * BN * NH;        // 4*1024*16
    float* attg   = att2   + (size_t)4 * BN * NH;        // 64

    hipLaunchKernelGGL(attg_kernel, dim3(1), dim3(64), 0, stream,
                       graph, ag_w, ag_b, attg);
    hipLaunchKernelGGL(proj_kernel, dim3(4 * 64), dim3(512), 0, stream,
                       node, m_w, m_b, sk_w, sk_b, a1_w, a1_b, a2_w, a2_b,
                       values, skipv, att1, att2);
    hipLaunchKernelGGL(gat_fused, dim3(4 * 64), dim3(512), 0, stream,
                       edge, adj, ae_w, ae_b, values, skipv, att1, att2, attg,
                       ln_s, ln_o, out);
}